// PhaseEncoding_46918222742189
// MI455X (gfx1250) — compile-verified
//
#include <hip/hip_runtime.h>

typedef __attribute__((ext_vector_type(2))) float v2f;
typedef __attribute__((ext_vector_type(8))) float v8f;

#define B_ 16
#define F_ 9
#define S_ 4096
#define D_ 512

// out[b,d,s] = x[b,d,s] + sum_f P[b,f,s] * W[f,d]
// Per-batch GEMM: W^T (D x F) @ P[b] (F x S).
// K=9 split as 8 (2 x V_WMMA_F32_16X16X4_F32, no padding/predication) + 1 (rank-1 FMA tail:
// all 8 C elements of a lane share the same column s, so tail = w8[v] * P[b,8,s]).
// One wave owns a 16(d) x 256(s) strip: A regs + tail weights loaded once, 16 s-tiles streamed.
__global__ __launch_bounds__(256) void phase_enc_wmma_kernel(
    const float* __restrict__ x,
    const float* __restrict__ P,
    const float* __restrict__ W,
    float* __restrict__ out)
{
    const int wave = (blockIdx.x * blockDim.x + threadIdx.x) >> 5;
    const int lane = threadIdx.x & 31;
    const int half = lane >> 4;   // which 16-lane half
    const int lid  = lane & 15;

    // wave -> (b, dtile, schunk): 16 * 32 * 16 = 8192 waves total
    const int schunk = wave & 15;
    const int dtile  = (wave >> 4) & 31;
    const int b      = wave >> 9;

    // ---- A tiles (W^T, 16 d x 4 f), ISA layout: M = lid, K = v + 2*half + 4*k; f = 0..7 ----
    const int d = dtile * 16 + lid;
    v2f a0, a1;
    {
        const int f0 = 2 * half;          // k = 0: f in {0,1} / {2,3}
        a0.x = W[f0       * D_ + d];
        a0.y = W[(f0 + 1) * D_ + d];
        const int f1 = 4 + 2 * half;      // k = 1: f in {4,5} / {6,7}
        a1.x = W[f1       * D_ + d];
        a1.y = W[(f1 + 1) * D_ + d];
    }

    // ---- f = 8 tail weights for this lane's 8 output rows (loop-invariant) ----
    const int drow0 = dtile * 16 + 8 * half;  // C/D layout: d = v + 8*half
    const float4* w8p = (const float4*)(W + 8 * D_ + drow0);  // 32B-aligned
    const float4 w8lo = w8p[0];
    const float4 w8hi = w8p[1];
    const float  w8[8] = { w8lo.x, w8lo.y, w8lo.z, w8lo.w,
                           w8hi.x, w8hi.y, w8hi.z, w8hi.w };

    const float* Pb    = P + b * (F_ * S_);
    const int    xbase = b * (D_ * S_);

    int s = schunk * 256 + lid;  // per-lane s column (N = lid)
#pragma unroll 2
    for (int t = 0; t < 16; ++t, s += 16) {
        // ---- B tiles (P, 4 f x 16 s), same lane/K striping as A; all f < 8, no predication ----
        v2f b0, b1;
        {
            const int f0 = 2 * half;
            b0.x = Pb[f0       * S_ + s];
            b0.y = Pb[(f0 + 1) * S_ + s];
            const int f1 = 4 + 2 * half;
            b1.x = Pb[f1       * S_ + s];
            b1.y = Pb[(f1 + 1) * S_ + s];
        }
        const float p8 = Pb[8 * S_ + s];   // f = 8 column value (shared by all 8 C elems)

        v8f c = {};
        c = __builtin_amdgcn_wmma_f32_16x16x4_f32(false, a0, false, b0, (short)0, c, false, false);
        c = __builtin_amdgcn_wmma_f32_16x16x4_f32(false, a1, false, b1, (short)0, c, false, false);

        // ---- epilogue: out = x + (acc + w8*p8) for the 16x16 tile ----
#pragma unroll
        for (int v = 0; v < 8; ++v) {
            const int idx = xbase + (drow0 + v) * S_ + s;
            out[idx] = x[idx] + fmaf(w8[v], p8, c[v]);
        }
    }
}

extern "C" void kernel_launch(void* const* d_in, const int* in_sizes, int n_in,
                              void* d_out, int out_size, void* d_ws, size_t ws_size,
                              hipStream_t stream) {
    const float* x = (const float*)d_in[0];  // (B, D, S)
    const float* P = (const float*)d_in[1];  // (B, F, S)
    const float* W = (const float*)d_in[2];  // (F, D)
    float* out = (float*)d_out;              // (B, D, S)

    // 8192 waves = 262144 threads = 1024 blocks x 256
    dim3 grid(1024), block(256);
    hipLaunchKernelGGL(phase_enc_wmma_kernel, grid, block, 0, stream, x, P, W, out);
}